// SimpleGAT_360777253388
// MI455X (gfx1250) — compile-verified
//
#include <hip/hip_runtime.h>
#include <hip/hip_bf16.h>

typedef __attribute__((ext_vector_type(2))) float v2f;
typedef __attribute__((ext_vector_type(8))) float v8f;

#define NEG_SLOPE 0.2f

// ---------------- utility: zero fill ----------------
__global__ void k_zero(float* p, long long n) {
    long long i = (long long)blockIdx.x * blockDim.x + threadIdx.x;
    if (i < n) p[i] = 0.0f;
}

// ------------- order-preserving float<->uint for atomicMax -------------
__device__ __forceinline__ unsigned fenc(float f) {
    unsigned b = __float_as_uint(f);
    return (b & 0x80000000u) ? ~b : (b | 0x80000000u);
}
__device__ __forceinline__ float fdec(unsigned u) {
    unsigned b = (u & 0x80000000u) ? (u & 0x7fffffffu) : ~u;
    return __uint_as_float(b);
}

// ---------------- WMMA GEMM: hp[N,64] = H[N,K] @ W[K,64] ----------------
// One block = 16 node rows, 128 threads = 4 waves, wave w owns output cols [16w,16w+16).
// Uses V_WMMA_F32_16X16X4_F32 (exact f32, matches reference precision).
// A layout (16x4 f32): lanes0-15 M=lane; v0 holds K = k0+(lane<16?0:2), v1 holds K+1.
// B layout (4x16 f32): v0: lanes0-15 K=k0, lanes16-31 K=k0+2 ; v1: +1. N = lane&15.
// D layout: VGPR r -> M = r + (lane<16?0:8), N = lane&15.
template<int K, bool CONCAT>
__global__ void k_gemm(const float* __restrict__ pos, const float* __restrict__ x,
                       const float* __restrict__ hdense, const float* __restrict__ W,
                       float* __restrict__ hp, int N) {
    const int lane = threadIdx.x & 31;
    const int wv   = threadIdx.x >> 5;          // 0..3
    const int base = blockIdx.x * 16;
    int mr = base + (lane & 15);
    if (mr >= N) mr = N - 1;                    // clamp (stores guarded below)
    const int koff = (lane < 16) ? 0 : 2;
    const int ncol = 16 * wv + (lane & 15);
    v8f c = {};
#pragma unroll
    for (int k0 = 0; k0 < K; k0 += 4) {
        const int ka = k0 + koff;
        float a0, a1;
        if (CONCAT) {  // H = [pos(2) | x(30)]
            a0 = (ka     < 2) ? pos[mr * 2 + ka]     : x[mr * 30 + (ka - 2)];
            a1 = ((ka+1) < 2) ? pos[mr * 2 + ka + 1] : x[mr * 30 + (ka - 1)];
        } else {
            const float* hr = hdense + (long long)mr * K;
            a0 = hr[ka]; a1 = hr[ka + 1];
        }
        v2f A = {a0, a1};
        v2f B = { W[ka * 64 + ncol], W[(ka + 1) * 64 + ncol] };
        c = __builtin_amdgcn_wmma_f32_16x16x4_f32(false, A, false, B,
                                                  (short)0, c, false, false);
    }
    const int moff = (lane < 16) ? 0 : 8;
#pragma unroll
    for (int r = 0; r < 8; ++r) {
        int row = base + r + moff;
        if (row < N) hp[(long long)row * 64 + ncol] = c[r];
    }
}

// ---------------- attention dot products: es[n]=hp[n]@a_src, ed[n]=hp[n]@a_dst --------
__global__ void k_attdots(const float* __restrict__ hp, const float* __restrict__ asrc,
                          const float* __restrict__ adst, float* __restrict__ es,
                          float* __restrict__ ed, int n) {
    int node = (int)(((long long)blockIdx.x * blockDim.x + threadIdx.x) >> 5);
    int lane = threadIdx.x & 31;
    if (node >= n) return;
    const float* hr = hp + (long long)node * 64;
    float p0 = hr[lane], p1 = hr[lane + 32];
    float s = p0 * asrc[lane] + p1 * asrc[lane + 32];
    float d = p0 * adst[lane] + p1 * adst[lane + 32];
#pragma unroll
    for (int off = 16; off; off >>= 1) {
        s += __shfl_xor(s, off, 32);
        d += __shfl_xor(d, off, 32);
    }
    if (lane == 0) { es[node] = s; ed[node] = d; }
}

// ---------------- edge pass 1: segment max (encoded uint atomicMax) ----------------
__global__ void k_edge_max(const int* __restrict__ ei, int E, int Nn,
                           const float* __restrict__ es, const float* __restrict__ ed,
                           unsigned* __restrict__ m) {
    int e = blockIdx.x * blockDim.x + threadIdx.x;
    const int Et = E + Nn;
    if (e >= Et) return;
    int j, i;
    if (e < E) { j = ei[e]; i = ei[E + e]; } else { j = i = e - E; }
    float v = es[j] + ed[i];
    v = (v > 0.0f) ? v : NEG_SLOPE * v;
    atomicMax(m + i, fenc(v));
}

// ---------------- edge pass 2: one wave per edge; weighted scatter-add ----------------
__global__ void k_edge_agg(const int* __restrict__ ei, int E, int Nn,
                           const float* __restrict__ es, const float* __restrict__ ed,
                           const unsigned* __restrict__ m, const float* __restrict__ hp,
                           float* __restrict__ denom, float* __restrict__ agg) {
    int e = (int)(((long long)blockIdx.x * blockDim.x + threadIdx.x) >> 5);
    int lane = threadIdx.x & 31;
    const int Et = E + Nn;
    if (e >= Et) return;
    int j, i;
    if (e < E) { j = ei[e]; i = ei[E + e]; } else { j = i = e - E; }
    float v = es[j] + ed[i];
    v = (v > 0.0f) ? v : NEG_SLOPE * v;
    float wgt = __expf(v - fdec(m[i]));
    if (lane == 0) atomicAdd(denom + i, wgt);
    const float* hr = hp + (long long)j * 64;
    float* ar = agg + (long long)i * 64;
    atomicAdd(ar + lane,      hr[lane]      * wgt);
    atomicAdd(ar + lane + 32, hr[lane + 32] * wgt);
}

// ---------------- normalize: h_out = agg/denom + bias ----------------
__global__ void k_norm(const float* __restrict__ agg, const float* __restrict__ denom,
                       const float* __restrict__ bias, float* __restrict__ hout, int n) {
    long long t = (long long)blockIdx.x * blockDim.x + threadIdx.x;
    if (t >= (long long)n * 64) return;
    int nn = (int)(t >> 6), c = (int)(t & 63);
    hout[t] = agg[t] / denom[nn] + bias[c];
}

// ---------------- global mean pool (sum + count via atomics; wave per node) ------------
__global__ void k_pool(const float* __restrict__ h, const int* __restrict__ batch,
                       float* __restrict__ pool, float* __restrict__ cnt, int n) {
    int node = (int)(((long long)blockIdx.x * blockDim.x + threadIdx.x) >> 5);
    int lane = threadIdx.x & 31;
    if (node >= n) return;
    int g = batch[node];
    atomicAdd(pool + g * 64 + lane,      h[(long long)node * 64 + lane]);
    atomicAdd(pool + g * 64 + lane + 32, h[(long long)node * 64 + lane + 32]);
    if (lane == 0) atomicAdd(cnt + g, 1.0f);
}

// ---------------- MLP head: mean -> relu(lin0) -> relu(lin1) ----------------
__global__ void k_head(const float* __restrict__ pool, const float* __restrict__ cnt,
                       const float* __restrict__ W0, const float* __restrict__ b0,
                       const float* __restrict__ W1, const float* __restrict__ b1,
                       float* __restrict__ out, int G) {
    int g = blockIdx.x * blockDim.x + threadIdx.x;
    if (g >= G) return;
    float inv = 1.0f / fmaxf(cnt[g], 1.0f);
    float mean[64];
#pragma unroll
    for (int c = 0; c < 64; ++c) mean[c] = pool[g * 64 + c] * inv;
    float mid[32];
#pragma unroll
    for (int mc = 0; mc < 32; ++mc) {
        float s = b0[mc];
        for (int k = 0; k < 64; ++k) s += mean[k] * W0[k * 32 + mc];
        mid[mc] = fmaxf(s, 0.0f);
    }
    for (int oc = 0; oc < 10; ++oc) {
        float s = b1[oc];
        for (int k = 0; k < 32; ++k) s += mid[k] * W1[k * 10 + oc];
        out[g * 10 + oc] = fmaxf(s, 0.0f);
    }
}

// ==========================================================================
extern "C" void kernel_launch(void* const* d_in, const int* in_sizes, int n_in,
                              void* d_out, int out_size, void* d_ws, size_t ws_size,
                              hipStream_t stream) {
    const float* x    = (const float*)d_in[0];
    const float* pos  = (const float*)d_in[1];
    const int*   ei   = (const int*)d_in[2];
    const int*   bat  = (const int*)d_in[3];
    const float* W0   = (const float*)d_in[4];
    const float* as0  = (const float*)d_in[5];
    const float* ad0  = (const float*)d_in[6];
    const float* bi0  = (const float*)d_in[7];
    const float* W1   = (const float*)d_in[8];
    const float* as1  = (const float*)d_in[9];
    const float* ad1  = (const float*)d_in[10];
    const float* bi1  = (const float*)d_in[11];
    const float* l0W  = (const float*)d_in[12];
    const float* l0b  = (const float*)d_in[13];
    const float* l1W  = (const float*)d_in[14];
    const float* l1b  = (const float*)d_in[15];
    float* out = (float*)d_out;

    const int N  = in_sizes[3];        // batch is [N]
    const int E  = in_sizes[2] / 2;    // edge_index is [2,E]
    const int G  = out_size / 10;
    const int Et = E + N;              // with self loops

    // ---- workspace layout (floats) ----
    float*    hp    = (float*)d_ws;                    // N*64
    float*    hbuf  = hp    + (long long)N * 64;       // N*64
    float*    agg   = hbuf  + (long long)N * 64;       // N*64  <-- zero region start
    float*    denom = agg   + (long long)N * 64;       // N
    unsigned* m     = (unsigned*)(denom + N);          // N
    float*    pool  = (float*)(m + N);                 // G*64
    float*    cnt   = pool  + (long long)G * 64;       // G     <-- zero region end
    float*    es    = cnt + G;                         // N
    float*    ed    = es + N;                          // N

    const long long zero_all  = (long long)N * 64 + 2LL * N + (long long)G * 64 + G;
    const long long zero_edge = (long long)N * 64 + 2LL * N;  // agg+denom+m only

    const int T = 256;
    dim3 blk(T);
    auto nb = [](long long work, int t) { return dim3((unsigned)((work + t - 1) / t)); };

    // zero agg/denom/m/pool/cnt
    k_zero<<<nb(zero_all, T), blk, 0, stream>>>(agg, zero_all);

    // ---- layer 0 ----
    k_gemm<32, true><<<dim3((N + 15) / 16), dim3(128), 0, stream>>>(pos, x, nullptr, W0, hp, N);
    k_attdots<<<nb((long long)N * 32, T), blk, 0, stream>>>(hp, as0, ad0, es, ed, N);
    k_edge_max<<<nb(Et, T), blk, 0, stream>>>(ei, E, N, es, ed, m);
    k_edge_agg<<<nb((long long)Et * 32, T), blk, 0, stream>>>(ei, E, N, es, ed, m, hp, denom, agg);
    k_norm<<<nb((long long)N * 64, T), blk, 0, stream>>>(agg, denom, bi0, hbuf, N);

    // re-zero agg/denom/m for layer 1
    k_zero<<<nb(zero_edge, T), blk, 0, stream>>>(agg, zero_edge);

    // ---- layer 1 ----
    k_gemm<64, false><<<dim3((N + 15) / 16), dim3(128), 0, stream>>>(nullptr, nullptr, hbuf, W1, hp, N);
    k_attdots<<<nb((long long)N * 32, T), blk, 0, stream>>>(hp, as1, ad1, es, ed, N);
    k_edge_max<<<nb(Et, T), blk, 0, stream>>>(ei, E, N, es, ed, m);
    k_edge_agg<<<nb((long long)Et * 32, T), blk, 0, stream>>>(ei, E, N, es, ed, m, hp, denom, agg);
    k_norm<<<nb((long long)N * 64, T), blk, 0, stream>>>(agg, denom, bi1, hbuf, N);

    // ---- pool + head ----
    k_pool<<<nb((long long)N * 32, T), blk, 0, stream>>>(hbuf, bat, pool, cnt, N);
    k_head<<<dim3(1), dim3(64), 0, stream>>>(pool, cnt, l0W, l0b, l1W, l1b, out, G);
}